// DIPModule_16690242912338
// MI455X (gfx1250) — compile-verified
//
#include <hip/hip_runtime.h>
#include <math.h>

// ---------------- problem constants (from reference) ----------------
#define IMG_W 512
#define IMG_H 512
#define HWPX  (IMG_W * IMG_H)     // 262144 px per plane
#define NPLANES 96                // B*C = 32*3
#define CHUNKS  32                // partial-sum chunks per plane (pass 1)
#define TH 8                      // output rows per block (pass 2)
#define TILE_ROWS (TH + 2)        // +2 halo rows
#define TILE_PITCH 520            // 4 pad | 512 data | 4 pad  (16B-aligned rows)
#define P1_BLOCK 256
#define P2_BLOCK 256

#ifndef __has_builtin
#define __has_builtin(x) 0
#endif
#if __has_builtin(__builtin_amdgcn_global_load_async_to_lds_b128)
#define USE_ASYNC_LDS 1
#endif

#define AS1 __attribute__((address_space(1)))
#define AS3 __attribute__((address_space(3)))
typedef int v4i __attribute__((ext_vector_type(4)));

__device__ __forceinline__ float clamp01(float v) {
    return fminf(fmaxf(v, 0.0f), 1.0f);
}
// x^g for x in [0,1): exp2(g*log2(x)) via v_log_f32 / v_exp_f32.
// log2(0) = -inf -> g*(-inf) = -inf -> exp2(-inf) = 0, matching 0^g = 0.
__device__ __forceinline__ float fast_pow(float x, float g) {
    return __builtin_amdgcn_exp2f(g * __builtin_amdgcn_logf(x));
}

// ---------------- pass 1a: per-(plane,chunk) partial sums of x^gamma ----------------
__global__ __launch_bounds__(P1_BLOCK)
void dip_pass1_partial(const float* __restrict__ x,
                       const float* __restrict__ gamma,
                       float* __restrict__ partials) {
    __shared__ float red[P1_BLOCK];
    const int chunk = blockIdx.x & (CHUNKS - 1);
    const int plane = blockIdx.x >> 5;          // / CHUNKS
    const float g = gamma[plane / 3];
    const float4* xv = reinterpret_cast<const float4*>(
        x + (size_t)plane * HWPX + (size_t)chunk * (HWPX / CHUNKS));
    const int t = threadIdx.x;

    float acc = 0.0f;
#pragma unroll
    for (int i = 0; i < (HWPX / CHUNKS) / 4 / P1_BLOCK; ++i) {  // 8 iters
        float4 v = xv[i * P1_BLOCK + t];
        acc += fast_pow(v.x, g) + fast_pow(v.y, g) +
               fast_pow(v.z, g) + fast_pow(v.w, g);
    }
    red[t] = acc;
    __syncthreads();
#pragma unroll
    for (int off = P1_BLOCK / 2; off > 0; off >>= 1) {
        if (t < off) red[t] += red[t + off];
        __syncthreads();
    }
    if (t == 0) partials[blockIdx.x] = red[0];
}

// ---------------- pass 1b: deterministic finalize -> mean per plane ----------------
__global__ void dip_pass1_final(const float* __restrict__ partials,
                                const float* __restrict__ wb,
                                float* __restrict__ means) {
    const int p = threadIdx.x;
    if (p < NPLANES) {
        float s = 0.0f;
        for (int j = 0; j < CHUNKS; ++j) s += partials[p * CHUNKS + j];   // fixed order
        means[p] = wb[p] * s * (1.0f / (float)HWPX);  // mean(pow(x,g)*wb) = wb*mean(pow)
    }
}

// ---------------- pass 2: transform + 3x3 sharpen + blend ----------------
__global__ __launch_bounds__(P2_BLOCK)
void dip_pass2_main(const float* __restrict__ x,
                    const float* __restrict__ gamma,
                    const float* __restrict__ wb,
                    const float* __restrict__ contrast,
                    const float* __restrict__ sharpen,
                    const float* __restrict__ means,
                    float* __restrict__ out) {
    __shared__ float tile[TILE_ROWS * TILE_PITCH];   // 20800 B

    const int t = threadIdx.x;
    const int rowBlock = blockIdx.x & 63;            // H/TH = 64
    const int plane    = blockIdx.x >> 6;
    const int b        = plane / 3;

    const float g   = gamma[b];
    const float w   = wb[plane];
    const float ctr = contrast[b];
    const float s   = sharpen[b];
    const float m   = means[plane];
    const int   r0  = rowBlock * TH;
    const float* xp = x + (size_t)plane * HWPX;

    // Zero the two pad columns the stencil touches (col 3 and col 516), all rows.
    if (t < 2 * TILE_ROWS) {
        int r   = t >> 1;
        int col = (t & 1) ? (4 + IMG_W) : 3;
        tile[r * TILE_PITCH + col] = 0.0f;
    }

    // ---- stage raw x rows [r0-1, r0+TH] into LDS (cols 4..515) ----
    // 10 rows * 512 floats = 1280 float4 groups, 5 per thread, coalesced 512B/wave.
#pragma unroll
    for (int j = 0; j < (TILE_ROWS * IMG_W) / (4 * P2_BLOCK); ++j) {   // 5 iters
        int q  = j * P2_BLOCK + t;       // float4 group id
        int r  = q >> 7;                 // 128 groups per row
        int c  = (q & 127) * 4;
        int gr = r0 - 1 + r;
        int lo = r * TILE_PITCH + 4 + c;
        if (gr >= 0 && gr < IMG_H) {
#ifdef USE_ASYNC_LDS
            __builtin_amdgcn_global_load_async_to_lds_b128(
                (AS1 v4i*)(xp + (size_t)gr * IMG_W + c),
                (AS3 v4i*)(&tile[lo]),
                /*offset=*/0, /*cpol=*/0);
#else
            float4 v = *reinterpret_cast<const float4*>(xp + (size_t)gr * IMG_W + c);
            *reinterpret_cast<float4*>(&tile[lo]) = v;
#endif
        } else {
            float4 z = make_float4(0.f, 0.f, 0.f, 0.f);
            *reinterpret_cast<float4*>(&tile[lo]) = z;   // conv zero-padding rows
        }
    }
#ifdef USE_ASYNC_LDS
#if __has_builtin(__builtin_amdgcn_s_wait_asynccnt)
    __builtin_amdgcn_s_wait_asynccnt(0);
#else
    asm volatile("s_wait_asynccnt 0" ::: "memory");
#endif
#endif
    __syncthreads();

    // ---- transform raw x -> u = clip((pow(x,g)*wb - m)*ctr + m, 0, 1) in place ----
    // Out-of-image halo rows must be u = 0 (conv pads u with zeros, not x).
#pragma unroll
    for (int j = 0; j < (TILE_ROWS * IMG_W) / P2_BLOCK; ++j) {  // 20 iters
        int q  = j * P2_BLOCK + t;
        int r  = q >> 9;
        int c  = q & (IMG_W - 1);
        int gr = r0 - 1 + r;
        int lo = r * TILE_PITCH + 4 + c;
        float v = tile[lo];
        float u = 0.0f;
        if (gr >= 0 && gr < IMG_H) {
            float p = fast_pow(v, g) * w;
            u = clamp01((p - m) * ctr + m);
        }
        tile[lo] = u;
    }
    __syncthreads();

    // ---- 3x3 sharpen + blend; scalar per-lane (consecutive lanes -> conflict-free LDS) ----
    float* outp = out + (size_t)plane * HWPX;
    const float onems = 1.0f - s;
#pragma unroll
    for (int k = 0; k < (TH * IMG_W) / P2_BLOCK; ++k) {   // 16 iters
        int q  = k * P2_BLOCK + t;
        int r  = q >> 9;                  // 0..7
        int c  = q & (IMG_W - 1);
        int bb = (r + 1) * TILE_PITCH + 4 + c;
        float uc = tile[bb];
        float s9 = tile[bb - TILE_PITCH - 1] + tile[bb - TILE_PITCH] + tile[bb - TILE_PITCH + 1]
                 + tile[bb - 1]              + uc                    + tile[bb + 1]
                 + tile[bb + TILE_PITCH - 1] + tile[bb + TILE_PITCH] + tile[bb + TILE_PITCH + 1];
        float sharp = 10.0f * uc - s9;    // 9*uc - (s9 - uc)
        float ov = clamp01(sharp * s + uc * onems);
        // NT store: keep x resident in the 192MB L2 for the whole dispatch
        __builtin_nontemporal_store(ov, outp + (size_t)(r0 + r) * IMG_W + c);
    }
}

// ---------------- host launch ----------------
extern "C" void kernel_launch(void* const* d_in, const int* in_sizes, int n_in,
                              void* d_out, int out_size, void* d_ws, size_t ws_size,
                              hipStream_t stream) {
    const float* x        = (const float*)d_in[0];
    const float* gamma    = (const float*)d_in[1];
    const float* wb       = (const float*)d_in[2];
    const float* contrast = (const float*)d_in[3];
    const float* sharpen  = (const float*)d_in[4];
    float* out = (float*)d_out;

    float* partials = (float*)d_ws;                  // 96*32 floats
    float* means    = partials + NPLANES * CHUNKS;   // 96 floats

    dip_pass1_partial<<<NPLANES * CHUNKS, P1_BLOCK, 0, stream>>>(x, gamma, partials);
    dip_pass1_final<<<1, 128, 0, stream>>>(partials, wb, means);
    dip_pass2_main<<<NPLANES * (IMG_H / TH), P2_BLOCK, 0, stream>>>(
        x, gamma, wb, contrast, sharpen, means, out);
}